// Local_APro_11871289606748
// MI455X (gfx1250) — compile-verified
//
#include <hip/hip_runtime.h>

#define BATCH 4
#define CH 3
#define H 384
#define W 384
#define K 5
#define K2 25
#define AFFP 28               // 25 weights padded to 28 floats (112B, 16B-aligned rows)
#define NPIX (BATCH * H * W)  // 589824
#define NUM_ITER 20

#define TW 32                 // prop tile width  (== blockDim.x, one wave row)
#define TH 8                  // prop tile height (== blockDim.y)
#define HW_ ((TW + 4) * (TH + 4))   // 36x12 halo tile = 432 floats

// LDS byte offset of a __shared__ element, via explicit generic->AS(3) cast
// (compiler folds to the workgroup-relative LDS address the async op needs).
__device__ __forceinline__ unsigned lds_offset(const float* p) {
    return (unsigned)(unsigned long long)(__attribute__((address_space(3))) const float*)p;
}

// ---------------------------------------------------------------------------
// Phase 1 (runs once): per-pixel normalized bilateral affinity + x0 = feat*mask
// grid (W/32, H/8, BATCH), block (32,8): no integer divisions needed.
// ---------------------------------------------------------------------------
__global__ __launch_bounds__(256) void aff_init_kernel(
    const float* __restrict__ img, const float* __restrict__ feat,
    const float* __restrict__ mask, float* __restrict__ aff,
    float* __restrict__ x0)
{
    const int b = blockIdx.z;
    const int x = blockIdx.x * TW + threadIdx.x;
    const int y = blockIdx.y * TH + threadIdx.y;

    const float* i0 = img + (size_t)(b * CH) * H * W;
    const float* i1 = i0 + (size_t)H * W;
    const float* i2 = i1 + (size_t)H * W;
    const int ci = y * W + x;
    // shift BEFORE zero-padded unfold (matches reference)
    const float c0 = i0[ci] + 10.f, c1 = i1[ci] + 10.f, c2 = i2[ci] + 10.f;

    const float invz2 = 1.0f / (0.15f * 0.15f);
    float a[K2];
    float sum = 0.f;
#pragma unroll
    for (int ky = 0; ky < K; ++ky) {
#pragma unroll
        for (int kx = 0; kx < K; ++kx) {
            int gy = y + ky - 2, gx = x + kx - 2;
            bool in = ((unsigned)gy < (unsigned)H) && ((unsigned)gx < (unsigned)W);
            int ni = gy * W + gx;
            float n0 = in ? i0[ni] + 10.f : 0.f;
            float n1 = in ? i1[ni] + 10.f : 0.f;
            float n2 = in ? i2[ni] + 10.f : 0.f;
            float d0 = n0 - c0, d1 = n1 - c1, d2 = n2 - c2;
            float dist = fmaf(d0, d0, fmaf(d1, d1, d2 * d2));
            float w = expf(-dist * invz2);
            a[ky * K + kx] = w;
            sum += w;
        }
    }
    const float inv = 1.0f / (sum + 1e-10f);

    float ap[AFFP];
#pragma unroll
    for (int k = 0; k < K2; ++k) ap[k] = a[k] * inv;
    ap[25] = ap[26] = ap[27] = 0.f;

    const size_t p = (size_t)b * H * W + ci;
    float4* row = (float4*)(aff + p * AFFP);
#pragma unroll
    for (int j = 0; j < 7; ++j)
        row[j] = make_float4(ap[4 * j], ap[4 * j + 1], ap[4 * j + 2], ap[4 * j + 3]);

    x0[p] = feat[p] * mask[p];
}

// ---------------------------------------------------------------------------
// Phase 2 (x20): one Jacobi step of the per-pixel-weighted 5x5 stencil.
// 32x8 output tile / block, 1 px/thread -> 2304 blocks (max TLP for the
// L2-latency-bound affinity stream). The 36x12 halo tile of x is staged into
// LDS with CDNA5 async global->LDS loads (ASYNCcnt); OOB slots are zeroed via
// plain DS stores (disjoint addresses -> no ASYNC/DS ordering hazard).
// ---------------------------------------------------------------------------
__global__ __launch_bounds__(256) void prop_kernel(
    const float* __restrict__ src, const float* __restrict__ aff,
    const float* __restrict__ mask, float* __restrict__ dst)
{
    __shared__ float xt[HW_];

    const int b   = blockIdx.z;
    const int tx0 = blockIdx.x * TW;
    const int ty0 = blockIdx.y * TH;
    const float* sp = src + (size_t)b * H * W;
    const int tid = threadIdx.y * TW + threadIdx.x;

    // ---- stage halo tile: async loads for in-bounds, zero-store for OOB ----
#pragma unroll
    for (int it = 0; it < (HW_ + 255) / 256; ++it) {
        int t = tid + it * 256;
        if (t < HW_) {
            int ly = t / (TW + 4), lx = t - ly * (TW + 4);
            int gy = ty0 + ly - 2, gx = tx0 + lx - 2;
            unsigned lds_b = lds_offset(&xt[t]);
            if (((unsigned)gy < (unsigned)H) && ((unsigned)gx < (unsigned)W)) {
                unsigned goff = (unsigned)((gy * W + gx) * 4);
                asm volatile("global_load_async_to_lds_b32 %0, %1, %2"
                             :: "v"(lds_b), "v"(goff), "s"(sp)
                             : "memory");
            } else {
                xt[t] = 0.f;
            }
        }
    }
    asm volatile("s_wait_asynccnt 0x0" ::: "memory");
    __syncthreads();

    // ---- compute: one output pixel per thread ----
    const int px = tx0 + threadIdx.x;
    const int py = ty0 + threadIdx.y;
    const size_t p = (size_t)b * H * W + (size_t)py * W + px;

    const float4* arow = (const float4*)(aff + p * AFFP);
    float a[AFFP];
#pragma unroll
    for (int j = 0; j < 7; ++j) {
        float4 v = arow[j];
        a[4 * j + 0] = v.x; a[4 * j + 1] = v.y;
        a[4 * j + 2] = v.z; a[4 * j + 3] = v.w;
    }

    const int lty = threadIdx.y + 2;
    const int ltx = threadIdx.x + 2;
    float acc = 0.f;
#pragma unroll
    for (int ky = 0; ky < K; ++ky)
#pragma unroll
        for (int kx = 0; kx < K; ++kx)
            acc = fmaf(a[ky * K + kx],
                       xt[(lty + ky - 2) * (TW + 4) + (ltx + kx - 2)], acc);

    dst[p] = acc * mask[p];
}

// ---------------------------------------------------------------------------
extern "C" void kernel_launch(void* const* d_in, const int* in_sizes, int n_in,
                              void* d_out, int out_size, void* d_ws, size_t ws_size,
                              hipStream_t stream)
{
    (void)in_sizes; (void)n_in; (void)out_size; (void)ws_size;

    const float* img  = (const float*)d_in[0];
    const float* feat = (const float*)d_in[1];
    const float* mask = (const float*)d_in[2];
    float* out = (float*)d_out;

    // workspace layout: aff (NPIX*28 f32, ~63MB) | xA | xB  (L2-resident: 192MB)
    float* aff = (float*)d_ws;
    float* xA  = aff + (size_t)NPIX * AFFP;
    float* xB  = xA + (size_t)NPIX;

    dim3 blk(TW, TH, 1);
    dim3 grd(W / TW, H / TH, BATCH);

    aff_init_kernel<<<grd, blk, 0, stream>>>(img, feat, mask, aff, xA);

    for (int i = 0; i < NUM_ITER; ++i) {
        const float* src = (i & 1) ? xB : xA;
        float* dst = (i == NUM_ITER - 1) ? out : ((i & 1) ? xA : xB);
        prop_kernel<<<grd, blk, 0, stream>>>(src, aff, mask, dst);
    }
}